// TPC_18322330485496
// MI455X (gfx1250) — compile-verified
//
#include <hip/hip_runtime.h>
#include <math.h>

typedef float v2f __attribute__((ext_vector_type(2)));
typedef float v8f __attribute__((ext_vector_type(8)));

#define NF 101
#define P 13
#define BT 240
#define BB 64
#define PS_STRIDE 218    // NF + 9*P
#define CUR_MAX_CH 2834  // 218*13 (even -> 1417 pairs)
#define XF_CH 220        // 202 + 16 flat + 2 zero pad (110 pairs)

#define WMMA_F32(acc, a, bb) \
  __builtin_amdgcn_wmma_f32_16x16x4_f32(false, (a), false, (bb), (short)0, (acc), false, false)

// ---------------------------------------------------------------------------
// Grouped dilated causal conv.  Input either plain (B,Cin,T) [layer 0: x] or
// channel-pair-interleaved (B,CP,T,2) [cur].  Output xt is pair-interleaved
// (B, G*6, T, 2): 6 b64 stores per thread.
// ---------------------------------------------------------------------------
template <int IPG, bool PIN>
__global__ __launch_bounds__(256) void conv_kernel(const float* __restrict__ in,
                                                   const float* __restrict__ w,
                                                   const float* __restrict__ bias,
                                                   float* __restrict__ out,
                                                   int G, int dil, int CPin) {
  const int g = blockIdx.x;
  const int b = blockIdx.y;
  __shared__ float sw[12 * IPG * 4];
  __shared__ float sb[12];
  const int nW = 12 * IPG * 4;
  const float* wg = w + (size_t)g * nW;
  for (int i = threadIdx.x; i < nW; i += 256) sw[i] = wg[i];
  if (threadIdx.x < 12) sb[threadIdx.x] = bias[g * 12 + threadIdx.x];
  __syncthreads();
  const int t = threadIdx.x;
  if (t >= BT) return;
  const int cc0 = g * IPG;
  float xv[IPG * 4];
#pragma unroll
  for (int ci = 0; ci < IPG; ++ci) {
#pragma unroll
    for (int kk = 0; kk < 4; ++kk) {
      const int tt = t - (3 - kk) * dil;
      float v = 0.0f;
      if (tt >= 0) {
        if (PIN) {
          const int cc = cc0 + ci;
          v = in[((size_t)b * CPin + (cc >> 1)) * (2 * BT) + 2 * tt + (cc & 1)];
        } else {
          v = in[((size_t)b * CPin + cc0 + ci) * BT + tt];
        }
      }
      xv[ci * 4 + kk] = v;
    }
  }
  float racc[12];
#pragma unroll
  for (int ko = 0; ko < 12; ++ko) {
    float acc = sb[ko];
#pragma unroll
    for (int q = 0; q < IPG * 4; ++q) acc = fmaf(sw[ko * IPG * 4 + q], xv[q], acc);
    racc[ko] = acc;
  }
  v2f* ob = (v2f*)out + ((size_t)b * (G * 6) + g * 6) * BT + t;
#pragma unroll
  for (int k2 = 0; k2 < 6; ++k2) {
    v2f v = {racc[2 * k2], racc[2 * k2 + 1]};
    ob[k2 * BT] = v;
  }
}

// ---------------------------------------------------------------------------
// Pointwise GEMM (V_WMMA_F32_16X16X4_F32); all loop bounds are compile-time
// constants (templated per layer) so every segment unrolls/pipelines.
// One wave = one 16-row M tile; Z selects the deterministic 2-way K split:
// Z=0 -> conv-out rows [0,S); Z=1 -> [S,F0) + linear(16) + xflat(220) + bias.
// Lane map (ISA 7.12.2): lanes<16 hold K=k,k+1; lanes>=16 hold K=k+2,k+3 ->
// pair-interleaved storage makes each fragment one aligned b64 load.
// ---------------------------------------------------------------------------
template <int F0, int S, int Z>
__global__ __launch_bounds__(128) void pointwise_kernel(
    const v2f* __restrict__ xt, const v2f* __restrict__ xpch,
    const v2f* __restrict__ xflat, const v2f* __restrict__ Wp,
    const float* __restrict__ bias, float* __restrict__ outp) {
  const int wave = blockIdx.x * 4 + (threadIdx.x >> 5);
  const int lane = threadIdx.x & 31;
  const int b = wave / 15;
  const int t0 = (wave % 15) * 16;
  const int n = lane & 15;
  const int kp = lane >> 4;  // pair offset 0/1
  const int t = t0 + (lane & 15);
  v8f acc = {};
  constexpr int k1lo = Z ? S : 0;
  constexpr int k1hi = Z ? F0 : S;

  const v2f* wp = Wp + ((size_t)(k1lo >> 1) + kp) * 16 + n;
  const v2f* a1 = xt + ((size_t)b * (F0 >> 1) + (k1lo >> 1) + kp) * BT + t;
#pragma unroll 8
  for (int k = k1lo; k < k1hi; k += 4) {
    acc = WMMA_F32(acc, a1[0], wp[0]);
    a1 += 2 * BT;
    wp += 32;
  }
  if (Z) {
    const v2f* wp2 = Wp + ((size_t)(F0 >> 1) + kp) * 16 + n;
    const v2f* a2 = xpch + ((size_t)b * 8 + kp) * BT + t;
#pragma unroll
    for (int k = 0; k < 16; k += 4) {
      acc = WMMA_F32(acc, a2[0], wp2[0]);
      a2 += 2 * BT;
      wp2 += 32;
    }
    const v2f* wp3 = Wp + ((size_t)((F0 + 16) >> 1) + kp) * 16 + n;
    const v2f* a3 = xflat + ((size_t)b * (XF_CH / 2) + kp) * BT + t;
#pragma unroll 8
    for (int k = 0; k < XF_CH; k += 4) {
      acc = WMMA_F32(acc, a3[0], wp3[0]);
      a3 += 2 * BT;
      wp3 += 32;
    }
  }
  const float bn = (Z && n < P) ? bias[n] : 0.0f;
  float* o = outp + (size_t)Z * (BB * BT * 16);
  const int half = (lane >> 4) << 3;
#pragma unroll
  for (int r = 0; r < 8; ++r) {
    const int row = t0 + r + half;
    o[((b * BT + row) << 4) + n] = acc[r] + bn;
  }
}

// Final GEMM: rows = [flat(16) | cur(2834 pad 2836)], Wp (2852/2 pairs x 32).
// Z=0 -> cur K [0,1416); Z=1 -> flat + cur K [1416,2836) + bias.
template <int Z>
__global__ __launch_bounds__(128) void los_kernel(const v2f* __restrict__ cur,
                                                  const float* __restrict__ flat,
                                                  const v2f* __restrict__ Wp,
                                                  const float* __restrict__ bias,
                                                  float* __restrict__ hidp) {
  const int gw = blockIdx.x * 4 + (threadIdx.x >> 5);
  const int nt = gw & 1;
  const int mt = gw >> 1;
  const int lane = threadIdx.x & 31;
  const int b = mt / 15;
  const int t0 = (mt % 15) * 16;
  const int t = t0 + (lane & 15);
  const int n = nt * 16 + (lane & 15);
  const int kp = lane >> 4;
  v8f acc = {};

  if (Z) {
    // flat segment: weight pair rows [0,8)
    const v2f* wp = Wp + (size_t)kp * 32 + n;
    const float* fb = flat + (b << 4) + 2 * kp;
#pragma unroll
    for (int k = 0; k < 16; k += 4) {
      v2f a = *(const v2f*)(fb + k);
      acc = WMMA_F32(acc, a, wp[0]);
      wp += 64;
    }
  }
  constexpr int klo = Z ? 1416 : 0;
  constexpr int khi = Z ? 2836 : 1416;
  const v2f* wp = Wp + ((size_t)((16 + klo) >> 1) + kp) * 32 + n;
  const v2f* a2 = cur + ((size_t)b * (CUR_MAX_CH / 2) + (klo >> 1) + kp) * BT + t;
#pragma unroll 8
  for (int k = klo; k < khi; k += 4) {
    acc = WMMA_F32(acc, a2[0], wp[0]);
    a2 += 2 * BT;
    wp += 64;
  }
  const float bn = (Z && n < 17) ? bias[n] : 0.0f;
  float* o = hidp + (size_t)Z * (BB * BT * 32);
  const int half = (lane >> 4) << 3;
#pragma unroll
  for (int r = 0; r < 8; ++r) {
    const int row = t0 + r + half;
    o[((b * BT + row) << 5) + n] = acc[r] + bn;
  }
}

__global__ __launch_bounds__(256) void combine_kernel(const float* __restrict__ p0,
                                                      const float* __restrict__ p1,
                                                      float* __restrict__ o, int nElem) {
  const int i = blockIdx.x * 256 + threadIdx.x;
  if (i < nElem) o[i] = p0[i] + p1[i];
}

// ---------------------------------------------------------------------------
// Data-movement / padding kernels
// ---------------------------------------------------------------------------
__global__ __launch_bounds__(256) void build_xflat_kernel(const float* __restrict__ x,
                                                          const float* __restrict__ flat,
                                                          float* __restrict__ xf) {
  const int c2 = blockIdx.x, b = blockIdx.y, t = threadIdx.x;
  if (t >= BT) return;
  float vv[2];
#pragma unroll
  for (int e = 0; e < 2; ++e) {
    const int c = 2 * c2 + e;
    float v = 0.0f;
    if (c < 2 * NF) v = x[(b * 2 * NF + c) * BT + t];
    else if (c < PS_STRIDE) v = flat[(b << 4) + (c - 2 * NF)];
    vv[e] = v;
  }
  v2f o = {vv[0], vv[1]};
  ((v2f*)xf)[((size_t)b * (XF_CH / 2) + c2) * BT + t] = o;
}

// Wpad, pair-interleaved: float at ((r>>1)*16 + n)*2 + (r&1).
__global__ __launch_bounds__(256) void build_wpad_pw_kernel(const float* __restrict__ W,
                                                            float* __restrict__ Wp,
                                                            int F0, int F1) {
  const int idx = blockIdx.x * 256 + threadIdx.x;
  const int Kpad = F0 + 16 + XF_CH;
  if (idx >= Kpad * 16) return;
  const int r = idx >> 4, nn = idx & 15;
  int ref = -1;
  if (r < F0) {
    ref = r;
  } else if (r < F0 + 16) {
    const int rr = r - F0;
    if (rr < F1) ref = F0 + rr;
  } else {
    const int rr = r - F0 - 16;
    if (rr < PS_STRIDE) ref = F0 + F1 + rr;
  }
  const float v = (ref >= 0 && nn < P) ? W[ref * P + nn] : 0.0f;
  Wp[(((r >> 1) * 16 + nn) << 1) + (r & 1)] = v;
}

__global__ __launch_bounds__(256) void build_wpad_los_kernel(const float* __restrict__ W,
                                                             float* __restrict__ Wp) {
  const int idx = blockIdx.x * 256 + threadIdx.x;
  if (idx >= 2852 * 32) return;
  const int r = idx >> 5, nn = idx & 31;
  const float v = (r < 2850 && nn < 17) ? W[r * 17 + nn] : 0.0f;
  Wp[(((r >> 1) * 32 + nn) << 1) + (r & 1)] = v;
}

__global__ __launch_bounds__(256) void append_kernel(const float* __restrict__ xp,
                                                     float* __restrict__ ps, int c0) {
  const int p = blockIdx.x, b = blockIdx.y, t = threadIdx.x;
  if (t >= BT) return;
  ps[(b * PS_STRIDE + c0 + p) * BT + t] = xp[((b * BT + t) << 4) + p];
}

__global__ __launch_bounds__(256) void append2_kernel(const float* __restrict__ xp,
                                                      float* __restrict__ xpch) {
  const int p2 = blockIdx.x, b = blockIdx.y, t = threadIdx.x;
  if (t >= BT) return;
  const int p0 = 2 * p2, p1 = p0 + 1;
  v2f o = {(p0 < P) ? xp[((b * BT + t) << 4) + p0] : 0.0f,
           (p1 < P) ? xp[((b * BT + t) << 4) + p1] : 0.0f};
  ((v2f*)xpch)[((size_t)b * 8 + p2) * BT + t] = o;
}

__global__ __launch_bounds__(256) void init_ps_kernel(const float* __restrict__ x,
                                                      float* __restrict__ ps) {
  const int c = blockIdx.x, b = blockIdx.y, t = threadIdx.x;
  if (t >= BT) return;
  ps[(b * PS_STRIDE + c) * BT + t] = x[(b * 2 * NF + c) * BT + t];
}

// Assemble next cur, pair-interleaved (B, CHp, T, 2); odd CH gets a 0 pad ch.
__global__ __launch_bounds__(256) void build_cur_kernel(const float* __restrict__ ps,
                                                        const float* __restrict__ xt,
                                                        const float* __restrict__ xp,
                                                        float* __restrict__ cur, int C) {
  const int cp = blockIdx.x, b = blockIdx.y, t = threadIdx.x;
  if (t >= BT) return;
  const int CH = (C + P) * 13;
  const int CHp = (CH + 1) >> 1;
  float vv[2];
#pragma unroll
  for (int e = 0; e < 2; ++e) {
    const int c = 2 * cp + e;
    float v = 0.0f;
    if (c < CH) {
      const int g = c / 13;
      const int j = c - g * 13;
      if (g < C) {
        if (j == 0) {
          v = ps[(b * PS_STRIDE + g) * BT + t];
        } else {
          const int cc = g * 12 + (j - 1);
          v = xt[((size_t)b * (C * 6) + (cc >> 1)) * (2 * BT) + 2 * t + (cc & 1)];
        }
      } else {
        v = xp[((b * BT + t) << 4) + (g - C)];
      }
    }
    vv[e] = v;
  }
  v2f o = {vv[0], vv[1]};
  ((v2f*)cur)[((size_t)b * CHp + cp) * BT + t] = o;
}

__global__ __launch_bounds__(256) void pred_kernel(const float* __restrict__ hidden,
                                                   const float* __restrict__ pw,
                                                   const float* __restrict__ pb,
                                                   float* __restrict__ out) {
  const int idx = blockIdx.x * 256 + threadIdx.x;
  if (idx >= BB * 235) return;
  const int b = idx / 235;
  const int t = (idx - b * 235) + 5;
  const float* h = hidden + ((b * BT + t) << 5);
  float s = pb[0];
#pragma unroll
  for (int k = 0; k < 17; ++k) s = fmaf(h[k], pw[k], s);
  float e = expf(s);
  out[idx] = fminf(fmaxf(e, 1.0f / 48.0f), 100.0f);
}

// ---------------------------------------------------------------------------
static constexpr int kG[9] = {101, 114, 127, 140, 153, 166, 179, 192, 205};

template <int I>
static void launch_pointwise(const v2f* xt, const v2f* xpch, const v2f* xfl,
                             const v2f* wpad, const float* bias, float* outp,
                             hipStream_t stream) {
  constexpr int F0 = (I == 0) ? 0 : kG[I - 1] * 12;
  constexpr int S = (F0 / 2) & ~3;
  pointwise_kernel<F0, S, 0><<<240, 128, 0, stream>>>(xt, xpch, xfl, wpad, bias, outp);
  pointwise_kernel<F0, S, 1><<<240, 128, 0, stream>>>(xt, xpch, xfl, wpad, bias, outp);
}

extern "C" void kernel_launch(void* const* d_in, const int* in_sizes, int n_in,
                              void* d_out, int out_size, void* d_ws, size_t ws_size,
                              hipStream_t stream) {
  const float* x = (const float*)d_in[0];
  const float* flat = (const float*)d_in[1];
  const float* conv_w[9];
  const float* conv_b[9];
  const float* pw_w[9];
  const float* pw_b[9];
  for (int i = 0; i < 9; ++i) {
    conv_w[i] = (const float*)d_in[2 + i];
    conv_b[i] = (const float*)d_in[11 + i];
    pw_w[i] = (const float*)d_in[20 + i];
    pw_b[i] = (const float*)d_in[29 + i];
  }
  const float* los_w = (const float*)d_in[38];
  const float* los_b = (const float*)d_in[39];
  const float* pred_w = (const float*)d_in[40];
  const float* pred_b = (const float*)d_in[41];

  auto al = [](size_t v) { return (v + 255) & ~(size_t)255; };
  char* base = (char*)d_ws;
  size_t off = 0;
  float* cur = (float*)(base + off);
  off += al((size_t)BB * CUR_MAX_CH * BT * 4 + 16384);  // slack for pad overread
  float* xt = (float*)(base + off);      off += al((size_t)BB * 2460 * BT * 4);
  float* ps = (float*)(base + off);      off += al((size_t)BB * PS_STRIDE * BT * 4);
  float* xfl = (float*)(base + off);     off += al((size_t)BB * XF_CH * BT * 4);
  float* xpch = (float*)(base + off);    off += al((size_t)BB * 16 * BT * 4);
  float* xpA = (float*)(base + off);     off += al((size_t)BB * BT * 16 * 4);
  float* xpB = (float*)(base + off);     off += al((size_t)BB * BT * 16 * 4);
  float* xpPart = (float*)(base + off);  off += al((size_t)2 * BB * BT * 16 * 4);
  float* hid = (float*)(base + off);     off += al((size_t)BB * BT * 32 * 4);
  float* hidPart = (float*)(base + off); off += al((size_t)2 * BB * BT * 32 * 4);
  float* wpad = (float*)(base + off);    off += al((size_t)2540 * 16 * 4);
  float* wpadL = (float*)(base + off);   off += al((size_t)2852 * 32 * 4);
  (void)ws_size; (void)in_sizes; (void)n_in; (void)out_size;

  hipMemsetAsync(xpch, 0, (size_t)BB * 16 * BT * 4, stream);  // layer-0 zeros
  build_xflat_kernel<<<dim3(XF_CH / 2, BB), 256, 0, stream>>>(x, flat, xfl);
  init_ps_kernel<<<dim3(NF, BB), 256, 0, stream>>>(x, ps);

  for (int i = 0; i < 9; ++i) {
    const int G = kG[i];
    float* xp_cur = (i & 1) ? xpB : xpA;
    const float* xp_prev = (i & 1) ? xpA : xpB;  // unused for i==0
    const int F0 = (i == 0) ? 0 : kG[i - 1] * 12;
    const int F1 = (i == 0) ? 0 : P;
    const int psOff = (i == 0) ? 0 : NF + P * (i - 1);
    const int Kpad = F0 + 16 + XF_CH;

    build_wpad_pw_kernel<<<(Kpad * 16 + 255) / 256, 256, 0, stream>>>(pw_w[i], wpad,
                                                                      F0, F1);
    if (i >= 1) {
      append_kernel<<<dim3(P, BB), 256, 0, stream>>>(xp_prev, ps, psOff);
      append2_kernel<<<dim3(8, BB), 256, 0, stream>>>(xp_prev, xpch);
    }
    switch (i) {
      case 0: launch_pointwise<0>((const v2f*)xt, (const v2f*)xpch, (const v2f*)xfl,
                                  (const v2f*)wpad, pw_b[i], xpPart, stream); break;
      case 1: launch_pointwise<1>((const v2f*)xt, (const v2f*)xpch, (const v2f*)xfl,
                                  (const v2f*)wpad, pw_b[i], xpPart, stream); break;
      case 2: launch_pointwise<2>((const v2f*)xt, (const v2f*)xpch, (const v2f*)xfl,
                                  (const v2f*)wpad, pw_b[i], xpPart, stream); break;
      case 3: launch_pointwise<3>((const v2f*)xt, (const v2f*)xpch, (const v2f*)xfl,
                                  (const v2f*)wpad, pw_b[i], xpPart, stream); break;
      case 4: launch_pointwise<4>((const v2f*)xt, (const v2f*)xpch, (const v2f*)xfl,
                                  (const v2f*)wpad, pw_b[i], xpPart, stream); break;
      case 5: launch_pointwise<5>((const v2f*)xt, (const v2f*)xpch, (const v2f*)xfl,
                                  (const v2f*)wpad, pw_b[i], xpPart, stream); break;
      case 6: launch_pointwise<6>((const v2f*)xt, (const v2f*)xpch, (const v2f*)xfl,
                                  (const v2f*)wpad, pw_b[i], xpPart, stream); break;
      case 7: launch_pointwise<7>((const v2f*)xt, (const v2f*)xpch, (const v2f*)xfl,
                                  (const v2f*)wpad, pw_b[i], xpPart, stream); break;
      case 8: launch_pointwise<8>((const v2f*)xt, (const v2f*)xpch, (const v2f*)xfl,
                                  (const v2f*)wpad, pw_b[i], xpPart, stream); break;
    }
    combine_kernel<<<(BB * BT * 16 + 255) / 256, 256, 0, stream>>>(
        xpPart, xpPart + (size_t)BB * BT * 16, xp_cur, BB * BT * 16);
    if (i == 0)
      conv_kernel<2, false><<<dim3(G, BB), 256, 0, stream>>>(x, conv_w[0], conv_b[0],
                                                             xt, G, 1, 2 * NF);
    else
      conv_kernel<13, true><<<dim3(G, BB), 256, 0, stream>>>(
          cur, conv_w[i], conv_b[i], xt, G, 3 * i, (G * 13 + 1) >> 1);
    const int CHp = ((G + P) * 13 + 1) >> 1;
    build_cur_kernel<<<dim3(CHp, BB), 256, 0, stream>>>(ps, xt, xp_cur, cur, G);
  }

  build_wpad_los_kernel<<<(2852 * 32 + 255) / 256, 256, 0, stream>>>(los_w, wpadL);
  los_kernel<0><<<480, 128, 0, stream>>>((const v2f*)cur, flat, (const v2f*)wpadL,
                                         los_b, hidPart);
  los_kernel<1><<<480, 128, 0, stream>>>((const v2f*)cur, flat, (const v2f*)wpadL,
                                         los_b, hidPart);
  combine_kernel<<<(BB * BT * 32 + 255) / 256, 256, 0, stream>>>(
      hidPart, hidPart + (size_t)BB * BT * 32, hid, BB * BT * 32);
  pred_kernel<<<(BB * 235 + 255) / 256, 256, 0, stream>>>(hid, pred_w, pred_b,
                                                          (float*)d_out);
}